// HiddenRepModel_81355270520882
// MI455X (gfx1250) — compile-verified
//
#include <hip/hip_runtime.h>

// ---------------------------------------------------------------------------
// Skip-gram w/ negative sampling + per-index MLP routing, tuned for MI455X:
//  * f32 WMMA (v_wmma_f32_16x16x4_f32) for the batched stoich->emb MLP
//  * weights packed once into WMMA B-operand layout, staged in LDS (317KB/WGP)
//  * wave32 tiles: each wave owns one 16-row tile; 8 waves (256 thr) per block
//  * no atomics; fixed-order reductions -> bitwise deterministic output
// HBM traffic ~0.3GB -> ~13us at 23.3TB/s; WMMA work ~42 GFLOP (negligible).
// ---------------------------------------------------------------------------

#define EMB_SZ   100000
#define STO_SZ   50000
#define S_DIM    100
#define D_DIM    128
#define H_DIM    256
#define BATCH    16384
#define KNEG     20

typedef __attribute__((ext_vector_type(2))) float v2f;
typedef __attribute__((ext_vector_type(8))) float v8f;

// LDS layout (bytes):
//   [0,       102400) packed W1  : float2[50][256]  (K-pairs x H)
//   [102400,  233472) packed W2  : float2[128][128] (K-pairs x D)
//   [233472 + wave*10496, +10496) per-wave region:
//        [0,4096)      hbuf   f32[16][64]   (layer1->layer2 chunk)
//        [4096,10496)  xbuf   f32[16][100]  (staged stoich tile)
//        [0,8192)      rowbuf f32[16][128]  (final rows; aliases hbuf/xbuf,
//                                            written only after compute)
#define LDS_W1_BYTES   102400
#define LDS_W2_BYTES   131072
#define LDS_WAVE_BYTES 10496
#define LDS_TOTAL      (LDS_W1_BYTES + LDS_W2_BYTES + 8 * LDS_WAVE_BYTES) // 317440

static __device__ __forceinline__ v8f wmma_f32(v2f a, v2f b, v8f c) {
  // 8-arg pattern per probe: (neg_a, A, neg_b, B, c_mod, C, reuse_a, reuse_b)
  return __builtin_amdgcn_wmma_f32_16x16x4_f32(false, a, false, b,
                                               (short)0, c, false, false);
}

static __device__ __forceinline__ float softplus_f(float x) {
  // stable log(1+exp(x))
  return fmaxf(x, 0.0f) + log1pf(expf(-fabsf(x)));
}

// Compute one 16-row tile of embeddings into rowbuf (f32[16][128]).
// MLP is evaluated for all 16 rows with WMMA (clamped stoich index, matching
// the reference's clip), then word rows are overwritten by table gathers.
static __device__ __forceinline__ void tile_embed(
    const int* __restrict__ idx_arr, int row_base,
    const float* __restrict__ stoich,
    const float* __restrict__ table,
    const float* __restrict__ b1g,
    const float* __restrict__ b2g,
    const float* __restrict__ lw1,    // LDS, packed float2 as floats
    const float* __restrict__ lw2,    // LDS, packed float2 as floats
    float* __restrict__ hbuf,
    float* __restrict__ xbuf,
    float* __restrict__ rowbuf) {
  const int lane = threadIdx.x & 31;
  const int nl   = lane & 15;
  const int hi   = lane >> 4;

  // Stage X tile (16 rows x 100 f32) into per-wave LDS.
  for (int i = lane; i < 16 * S_DIM; i += 32) {
    int m = i / S_DIM;
    int e = i - m * S_DIM;
    int idx  = idx_arr[row_base + m];
    int sidx = idx - EMB_SZ;
    sidx = sidx < 0 ? 0 : (sidx > STO_SZ - 1 ? STO_SZ - 1 : sidx);
    xbuf[i] = stoich[(long)sidx * S_DIM + e];
  }
  // per-wave buffer; same-wave LDS RAW ordering handled via s_wait_dscnt.

  // Layer-2 accumulators: 16x128 out tile, bias-initialized.
  v8f acc2[8];
  for (int t = 0; t < 8; ++t) {
    float bv = b2g[16 * t + nl];
    for (int j = 0; j < 8; ++j) acc2[t][j] = bv;
  }

  const float* xrow = xbuf + nl * S_DIM;  // A-operand row for this lane
  for (int c = 0; c < 4; ++c) {
    // ---- layer 1, H columns [64c, 64c+64) -------------------------------
    v8f h[4];
    for (int t = 0; t < 4; ++t) {
      float bv = b1g[64 * c + 16 * t + nl];
      for (int j = 0; j < 8; ++j) h[t][j] = bv;
    }
    for (int kk = 0; kk < 25; ++kk) {          // K = 100 = 25 x 4
      v2f a = *(const v2f*)(xrow + 4 * kk + 2 * hi);
      const float* brow = lw1 + ((2 * kk + hi) * 256 + 64 * c + nl) * 2;
#pragma unroll
      for (int t = 0; t < 4; ++t) {
        v2f b = *(const v2f*)(brow + 32 * t);  // float2 stride 16 -> 32 floats
        h[t] = wmma_f32(a, b, h[t]);
      }
    }
    // ReLU + stage to hbuf (local cols 0..63)
    for (int t = 0; t < 4; ++t)
      for (int vg = 0; vg < 8; ++vg)
        hbuf[(vg + 8 * hi) * 64 + 16 * t + nl] = fmaxf(h[t][vg], 0.0f);

    // ---- layer 2 partial over this 64-wide K chunk ----------------------
    for (int kk = 0; kk < 16; ++kk) {
      v2f a2 = *(const v2f*)(hbuf + nl * 64 + 4 * kk + 2 * hi);
      const float* brow = lw2 + ((32 * c + 2 * kk + hi) * 128 + nl) * 2;
#pragma unroll
      for (int t = 0; t < 8; ++t) {
        v2f b = *(const v2f*)(brow + 32 * t);
        acc2[t] = wmma_f32(a2, b, acc2[t]);
      }
    }
  }

  // Spill MLP result: lane holds out[m = vg+8*hi][d = 16t+nl].
  for (int t = 0; t < 8; ++t)
    for (int vg = 0; vg < 8; ++vg)
      rowbuf[(vg + 8 * hi) * D_DIM + 16 * t + nl] = acc2[t][vg];

  // Overwrite word rows with table gathers (32 lanes x float4 = 1 row).
  for (int m = 0; m < 16; ++m) {
    int idx = idx_arr[row_base + m];
    if (idx < EMB_SZ) {
      float4 src = ((const float4*)(table + (long)idx * D_DIM))[lane];
      ((float4*)(rowbuf + m * D_DIM))[lane] = src;
    }
  }
}

// Cooperative copy of packed weight images (global ws -> LDS), then barrier.
static __device__ __forceinline__ void stage_weights(
    const float* __restrict__ gw1, const float* __restrict__ gw2,
    float* __restrict__ lw1, float* __restrict__ lw2) {
  const int tid = threadIdx.x;
  for (int i = tid; i < LDS_W1_BYTES / 16; i += 256)
    ((float4*)lw1)[i] = ((const float4*)gw1)[i];
  for (int i = tid; i < LDS_W2_BYTES / 16; i += 256)
    ((float4*)lw2)[i] = ((const float4*)gw2)[i];
  __syncthreads();
}

// ---- weight packing: W1[h][s] -> float2[(s/2)][h], W2[d][h] -> float2[(h/2)][d]
__global__ void __launch_bounds__(256) pack_w1_kernel(const float* __restrict__ w1,
                                                      float* __restrict__ dst) {
  int e = blockIdx.x * 256 + threadIdx.x;          // 50*256 = 12800 float2
  int k2 = e >> 8, h = e & 255;
  dst[2 * e + 0] = w1[h * S_DIM + 2 * k2 + 0];
  dst[2 * e + 1] = w1[h * S_DIM + 2 * k2 + 1];
}

__global__ void __launch_bounds__(256) pack_w2_kernel(const float* __restrict__ w2,
                                                      float* __restrict__ dst) {
  int e = blockIdx.x * 256 + threadIdx.x;          // 128*128 = 16384 float2
  int k2 = e >> 7, d = e & 127;
  dst[2 * e + 0] = w2[d * H_DIM + 2 * k2 + 0];
  dst[2 * e + 1] = w2[d * H_DIM + 2 * k2 + 1];
}

// ---- pass 1: emb_u (blocks 0..127) and emb_v (blocks 128..255) -------------
__global__ void __launch_bounds__(256) embed_kernel(
    const float* __restrict__ u_emb, const float* __restrict__ v_emb,
    const float* __restrict__ t_b1, const float* __restrict__ t_b2,
    const float* __restrict__ c_b1, const float* __restrict__ c_b2,
    const float* __restrict__ stoich,
    const int* __restrict__ pos_u, const int* __restrict__ pos_v,
    const float* __restrict__ w1p_t, const float* __restrict__ w2p_t,
    const float* __restrict__ w1p_c, const float* __restrict__ w2p_c,
    float* __restrict__ emb_u, float* __restrict__ emb_v) {
  extern __shared__ char smem[];
  float* lw1 = (float*)(smem);
  float* lw2 = (float*)(smem + LDS_W1_BYTES);
  const int side = blockIdx.x >> 7;   // 0 = U-side, 1 = V-side
  stage_weights(side ? w1p_c : w1p_t, side ? w2p_c : w2p_t, lw1, lw2);

  const int wave = threadIdx.x >> 5;
  const int lane = threadIdx.x & 31;
  float* wreg   = (float*)(smem + LDS_W1_BYTES + LDS_W2_BYTES + wave * LDS_WAVE_BYTES);
  float* hbuf   = wreg;
  float* xbuf   = wreg + 1024;
  float* rowbuf = wreg;

  const int*   idx_arr = side ? pos_v : pos_u;
  const float* table   = side ? v_emb : u_emb;
  const float* b1g     = side ? c_b1 : t_b1;
  const float* b2g     = side ? c_b2 : t_b2;
  float*       outp    = side ? emb_v : emb_u;
  const int row_base = (blockIdx.x & 127) * 128 + wave * 16;

  tile_embed(idx_arr, row_base, stoich, table, b1g, b2g, lw1, lw2,
             hbuf, xbuf, rowbuf);

  for (int m = 0; m < 16; ++m)
    ((float4*)(outp + (long)(row_base + m) * D_DIM))[lane] =
        ((const float4*)(rowbuf + m * D_DIM))[lane];
}

// ---- pass 2: neg rows -> embedding tile -> dot with emb_u[b] -> softplus ----
__global__ void __launch_bounds__(256) neg_kernel(
    const float* __restrict__ v_emb,
    const float* __restrict__ c_b1, const float* __restrict__ c_b2,
    const float* __restrict__ stoich, const int* __restrict__ neg_v,
    const float* __restrict__ w1p_c, const float* __restrict__ w2p_c,
    const float* __restrict__ emb_u, float* __restrict__ neg_s) {
  extern __shared__ char smem[];
  float* lw1 = (float*)(smem);
  float* lw2 = (float*)(smem + LDS_W1_BYTES);
  stage_weights(w1p_c, w2p_c, lw1, lw2);

  const int wave = threadIdx.x >> 5;
  const int lane = threadIdx.x & 31;
  float* wreg   = (float*)(smem + LDS_W1_BYTES + LDS_W2_BYTES + wave * LDS_WAVE_BYTES);
  float* hbuf   = wreg;
  float* xbuf   = wreg + 1024;
  float* rowbuf = wreg;

  const int row_base = blockIdx.x * 128 + wave * 16;
  tile_embed(neg_v, row_base, stoich, v_emb, c_b1, c_b2, lw1, lw2,
             hbuf, xbuf, rowbuf);

  for (int m = 0; m < 16; ++m) {
    const int row = row_base + m;
    const int b   = row / KNEG;
    float4 e = ((const float4*)(rowbuf + m * D_DIM))[lane];
    float4 u = ((const float4*)(emb_u + (long)b * D_DIM))[lane];
    float p = e.x * u.x + e.y * u.y + e.z * u.z + e.w * u.w;
    for (int off = 16; off > 0; off >>= 1) p += __shfl_xor(p, off, 32);
    if (lane == 0) {
      float s = fminf(fmaxf(p, -10.0f), 10.0f);
      neg_s[row] = softplus_f(s);   // = -log_sigmoid(-s)
    }
  }
}

// ---- pass 3: per-b score + block tree reduction (fixed order) ---------------
__global__ void __launch_bounds__(256) score_kernel(
    const float* __restrict__ emb_u, const float* __restrict__ emb_v,
    const float* __restrict__ neg_s, float* __restrict__ partial) {
  const int tid = threadIdx.x;
  const int b   = blockIdx.x * 256 + tid;
  const float4* pu = (const float4*)(emb_u + (long)b * D_DIM);
  const float4* pv = (const float4*)(emb_v + (long)b * D_DIM);
  float dot = 0.0f;
  for (int i = 0; i < D_DIM / 4; ++i) {
    float4 a = pu[i], c = pv[i];
    dot += a.x * c.x + a.y * c.y + a.z * c.z + a.w * c.w;
  }
  float s   = fminf(fmaxf(dot, -10.0f), 10.0f);
  float tot = softplus_f(-s);      // = -log_sigmoid(s)
  for (int k = 0; k < KNEG; ++k) tot += neg_s[b * KNEG + k];

  __shared__ float red[256];
  red[tid] = tot;
  __syncthreads();
  for (int st = 128; st > 0; st >>= 1) {
    if (tid < st) red[tid] += red[tid + st];
    __syncthreads();
  }
  if (tid == 0) partial[blockIdx.x] = red[0];
}

__global__ void final_kernel(const float* __restrict__ partial,
                             float* __restrict__ out) {
  if (threadIdx.x == 0 && blockIdx.x == 0) {
    float s = 0.0f;
    for (int i = 0; i < 64; ++i) s += partial[i];   // fixed order
    out[0] = s / (float)BATCH;
  }
}

extern "C" void kernel_launch(void* const* d_in, const int* in_sizes, int n_in,
                              void* d_out, int out_size, void* d_ws, size_t ws_size,
                              hipStream_t stream) {
  (void)in_sizes; (void)n_in; (void)out_size; (void)ws_size;
  const float* u_emb  = (const float*)d_in[0];
  const float* v_emb  = (const float*)d_in[1];
  const float* t_w1   = (const float*)d_in[2];
  const float* t_b1   = (const float*)d_in[3];
  const float* t_w2   = (const float*)d_in[4];
  const float* t_b2   = (const float*)d_in[5];
  const float* c_w1   = (const float*)d_in[6];
  const float* c_b1   = (const float*)d_in[7];
  const float* c_w2   = (const float*)d_in[8];
  const float* c_b2   = (const float*)d_in[9];
  const float* stoich = (const float*)d_in[10];
  const int*   pos_u  = (const int*)d_in[11];
  const int*   pos_v  = (const int*)d_in[12];
  const int*   neg_v  = (const int*)d_in[13];

  // workspace layout (floats); total ~4.64M floats (~18.6 MB)
  float* ws      = (float*)d_ws;
  float* emb_u   = ws;                       // 16384*128
  float* emb_v   = emb_u + (long)BATCH * D_DIM;
  float* neg_s   = emb_v + (long)BATCH * D_DIM;      // 16384*20
  float* partial = neg_s + (long)BATCH * KNEG;       // 64
  float* w1p_t   = partial + 64;             // 25600 floats (float2[50][256])
  float* w2p_t   = w1p_t + 25600;            // 32768 floats (float2[128][128])
  float* w1p_c   = w2p_t + 32768;
  float* w2p_c   = w1p_c + 25600;

  pack_w1_kernel<<<50, 256, 0, stream>>>(t_w1, w1p_t);
  pack_w2_kernel<<<64, 256, 0, stream>>>(t_w2, w2p_t);
  pack_w1_kernel<<<50, 256, 0, stream>>>(c_w1, w1p_c);
  pack_w2_kernel<<<64, 256, 0, stream>>>(c_w2, w2p_c);

  embed_kernel<<<256, 256, LDS_TOTAL, stream>>>(
      u_emb, v_emb, t_b1, t_b2, c_b1, c_b2, stoich, pos_u, pos_v,
      w1p_t, w2p_t, w1p_c, w2p_c, emb_u, emb_v);

  neg_kernel<<<(BATCH * KNEG) / 128, 256, LDS_TOTAL, stream>>>(
      v_emb, c_b1, c_b2, stoich, neg_v, w1p_c, w2p_c, emb_u, neg_s);

  score_kernel<<<BATCH / 256, 256, 0, stream>>>(emb_u, emb_v, neg_s, partial);
  final_kernel<<<1, 32, 0, stream>>>(partial, (float*)d_out);
}